// VolumeAwareLoss_49211735277884
// MI455X (gfx1250) — compile-verified
//
#include <hip/hip_runtime.h>
#include <hip/hip_bf16.h>
#include <math.h>

#define DEV __device__ __forceinline__

typedef float v2f __attribute__((ext_vector_type(2)));
typedef float v8f __attribute__((ext_vector_type(8)));

namespace {
constexpr int Bb = 2, Cc = 5, Dd = 96, Hh = 96, Ww = 96;
constexpr int DHW  = Dd * Hh * Ww;      // 884736
constexpr int ROWS = Dd * Hh;           // 9216 rows per batch
constexpr int NPAIR = Bb * (Cc - 1);    // 8 surface pairs
constexpr int WPP  = ROWS * 3;          // bit-mask words per pair (96 bits = 3 u32 per row)
constexpr float EPSF = 1e-5f;
constexpr float INFF = 1.0e9f;
constexpr float ALPHA = 0.3f, BETA = 0.7f;

// accumulator slots (floats; integer slots reinterpreted)
constexpr int AVOLS = 0;    // 10: target sums  [b*5+c]
constexpr int AINTER = 10;  // 8:  sum p*t      [b*4+c']
constexpr int APSUM = 18;   // 8:  sum p        [b*4+c']
constexpr int ACE   = 26;   // 2:  CE partial per b
constexpr int ASURF = 28;   // 8:  sum of sqrt(dist) over pred boundary
constexpr int ACNT  = 36;   // 8:  (u32) pred boundary voxel count
constexpr int APANY = 44;   // 8:  (u32) pred boundary any-flag
constexpr int ATANY = 52;   // 8:  (u32) target boundary any-flag
constexpr int ACC_SLOTS = 64;

// workspace layout (bytes)
constexpr int    NBLK1 = Bb * ROWS;                       // 18432 blocks in K1
constexpr int    SBLK  = DHW / 256;                       // 3456 surf blocks per pair
constexpr size_t ACC_OFF  = 0;
constexpr size_t P1_OFF   = 256;
constexpr size_t P2_OFF   = P1_OFF + (size_t)NBLK1 * 14 * 4;
constexpr size_t PM_OFF   = P2_OFF + (size_t)NPAIR * SBLK * 4;
constexpr size_t TM_OFF   = PM_OFF  + (size_t)NPAIR * WPP * 4;
constexpr size_t PMBD_OFF = TM_OFF  + (size_t)NPAIR * WPP * 4;
constexpr size_t TMBD_OFF = PMBD_OFF + (size_t)NPAIR * WPP * 4;
constexpr size_t F_OFF    = TMBD_OFF + (size_t)NPAIR * WPP * 4;   // + 8*884736*4 ≈ 27 MB
} // namespace

// ---------------------------------------------------------------------------
// wave32 full-lane sum using V_WMMA_F32_16X16X4_F32 as a reduction engine.
// A (16x4 f32, 2 VGPRs): lane L (0-15) supplies A[L][0]=v, A[L][1]=0;
// lane L+16 supplies A[L][2]=v, A[L][3]=0. B = ones(4x16).
// D[m][n] = v[m] + v[m+16]; lanes 0-15 hold rows 0-7, lanes 16-31 rows 8-15:
// sum of the 8 D VGPRs + one ds_swizzle XOR-16 swap = full 32-lane sum in
// every lane. EXEC must be all ones at the call site (callers use full waves).
// ---------------------------------------------------------------------------
DEV float half_swap_add(float t) {
  // ds_swizzle group-of-32: and=0x1F, or=0, xor=0x10  -> lane L reads L^16
  const int s = __builtin_amdgcn_ds_swizzle(__float_as_int(t), 0x401F);
  return t + __int_as_float(s);
}

DEV float wave_sum(float v) {
#if __has_builtin(__builtin_amdgcn_wmma_f32_16x16x4_f32)
  v2f a; a.x = v;    a.y = 0.0f;
  v2f b; b.x = 1.0f; b.y = 1.0f;
  v8f c = {};
  c = __builtin_amdgcn_wmma_f32_16x16x4_f32(false, a, false, b,
                                            (short)0, c, false, false);
  float t = c[0] + c[1] + c[2] + c[3] + c[4] + c[5] + c[6] + c[7];
  return half_swap_add(t);
#else
  for (int off = 16; off > 0; off >>= 1) v += __shfl_xor(v, off, 32);
  return v;
#endif
}

// Async global->LDS copy of one b32 per lane (CDNA5 GLOBAL_LOAD_ASYNC_TO_LDS,
// GV mode: 64-bit VGPR address, SADDR=off). lds_off = per-lane LDS byte addr
// (low 32 bits of the generic shared-space pointer). Tracked by ASYNCcnt.
DEV void async_load_lds_b32(unsigned lds_off, const float* gaddr) {
  asm volatile("global_load_async_to_lds_b32 %0, %1, off"
               :: "v"(lds_off), "v"(gaddr) : "memory");
}
DEV void wait_asynccnt0() {
  asm volatile("s_wait_asynccnt 0" ::: "memory");
}

// K0: zero the accumulator block
__global__ void k_init(float* __restrict__ acc) {
  if (threadIdx.x < ACC_SLOTS) acc[threadIdx.x] = 0.0f;
}

// K1: fused softmax / log-softmax / per-class sums / bit masks.
// grid = B*D*H blocks of 96 threads (one row, 3 full wave32s).
__global__ void k_fused(const float* __restrict__ pred,
                        const float* __restrict__ target,
                        unsigned* __restrict__ pm, unsigned* __restrict__ tm,
                        float* __restrict__ partials) {
  const int tid = threadIdx.x;              // 0..95 == w
  const int b   = blockIdx.x / ROWS;
  const int row = blockIdx.x % ROWS;        // z*H + y
  const size_t base = (size_t)b * Cc * DHW + (size_t)row * Ww + tid;

  float p[5], t[5];
#pragma unroll
  for (int c = 0; c < 5; ++c) {
    p[c] = pred[base + (size_t)c * DHW];
    t[c] = target[base + (size_t)c * DHW];
  }
  float m = p[0];
#pragma unroll
  for (int c = 1; c < 5; ++c) m = fmaxf(m, p[c]);
  float e[5], Z = 0.0f;
#pragma unroll
  for (int c = 0; c < 5; ++c) { e[c] = expf(p[c] - m); Z += e[c]; }
  const float invZ = 1.0f / Z;
  const float logZ = logf(Z);
  float prob[5], ce = 0.0f;
#pragma unroll
  for (int c = 0; c < 5; ++c) {
    prob[c] = e[c] * invZ;
    ce -= t[c] * ((p[c] - m) - logZ);
  }

  // bit-packed masks for classes 1..4 via LDS atomicOr (W=96 -> 3 words/row)
  __shared__ unsigned smask[24];            // [0..11]=pm, [12..23]=tm
  if (tid < 24) smask[tid] = 0u;
  __syncthreads();
  const int wword = tid >> 5, lanebit = tid & 31;
#pragma unroll
  for (int cp = 0; cp < 4; ++cp) {
    if (prob[cp + 1] > 0.5f) atomicOr(&smask[cp * 3 + wword], 1u << lanebit);
    if (t[cp + 1]    > 0.5f) atomicOr(&smask[12 + cp * 3 + wword], 1u << lanebit);
  }
  __syncthreads();
  if (tid < 12) {
    const int cp = tid / 3, wwi = tid % 3;
    const size_t mo = (size_t)(b * 4 + cp) * WPP + (size_t)row * 3 + wwi;
    pm[mo] = smask[tid];
    tm[mo] = smask[12 + tid];
  }

  // 14 per-row reductions: vols[5], inter[4], psum[4], ce
  float vals[14];
#pragma unroll
  for (int c = 0; c < 5; ++c) vals[c] = t[c];
#pragma unroll
  for (int cp = 0; cp < 4; ++cp) vals[5 + cp] = prob[cp + 1] * t[cp + 1];
#pragma unroll
  for (int cp = 0; cp < 4; ++cp) vals[9 + cp] = prob[cp + 1];
  vals[13] = ce;

  // WMMA wave reductions with EXEC all-ones; single divergent store region.
  float sv[14];
#pragma unroll
  for (int k = 0; k < 14; ++k) sv[k] = wave_sum(vals[k]);

  __shared__ float sred[3][14];
  const int wid = tid >> 5, lane = tid & 31;
  if (lane == 0) {
#pragma unroll
    for (int k = 0; k < 14; ++k) sred[wid][k] = sv[k];
  }
  __syncthreads();
  if (tid < 14)
    partials[(size_t)blockIdx.x * 14 + tid] = sred[0][tid] + sred[1][tid] + sred[2][tid];
}

// K2: deterministic second-stage reduction; grid = 2*14 blocks of 256.
__global__ void k_reduce_main(const float* __restrict__ partials,
                              float* __restrict__ acc) {
  const int b = blockIdx.x / 14, k = blockIdx.x % 14;
  const int tid = threadIdx.x;
  float s = 0.0f;
  for (int i = tid; i < ROWS; i += 256)
    s += partials[(size_t)(b * ROWS + i) * 14 + k];
  const float wsum = wave_sum(s);
  __shared__ float sl[8];
  if ((tid & 31) == 0) sl[tid >> 5] = wsum;
  __syncthreads();
  if (tid == 0) {
    float tot = 0.0f;
    for (int j = 0; j < 8; ++j) tot += sl[j];
    const int slot = (k < 5)  ? AVOLS + b * 5 + k
                   : (k < 9)  ? AINTER + b * 4 + (k - 5)
                   : (k < 13) ? APSUM + b * 4 + (k - 9)
                              : ACE + b;
    acc[slot] = tot;
  }
}

// K3: boundary = m & ~erode(m) on bit-packed words; integer atomics only.
__global__ void k_boundary(const unsigned* __restrict__ pm,
                           const unsigned* __restrict__ tm,
                           unsigned* __restrict__ pmbd,
                           unsigned* __restrict__ tmbd,
                           unsigned* __restrict__ accu) {
  const int idx  = blockIdx.x * 256 + threadIdx.x;     // < NPAIR*WPP (exact)
  const int pair = idx / WPP;
  const int r    = idx - pair * WPP;
  const int row  = r / 3, ww = r - row * 3;
  const int z = row / Hh, y = row - z * Hh;
  const size_t pbase = (size_t)pair * WPP;

  auto rd = [&](const unsigned* mbase, int zz, int yy, int w2) -> unsigned {
    if (zz < 0 || zz >= Dd || yy < 0 || yy >= Hh || w2 < 0 || w2 >= 3) return 0u;
    return mbase[pbase + (size_t)(zz * Hh + yy) * 3 + w2];
  };
  auto bnd_of = [&](const unsigned* mb) -> unsigned {
    const unsigned Cw = rd(mb, z, y, ww);
    const unsigned zm = rd(mb, z - 1, y, ww), zp = rd(mb, z + 1, y, ww);
    const unsigned ym = rd(mb, z, y - 1, ww), yp = rd(mb, z, y + 1, ww);
    const unsigned lw = rd(mb, z, y, ww - 1), rw = rd(mb, z, y, ww + 1);
    const unsigned west = (Cw << 1) | (lw >> 31);   // m[w-1]
    const unsigned east = (Cw >> 1) | (rw << 31);   // m[w+1]
    return Cw & ~(zm & zp & ym & yp & west & east);
  };
  const unsigned bp = bnd_of(pm);
  const unsigned bt = bnd_of(tm);
  pmbd[idx] = bp;
  tmbd[idx] = bt;

  // WPP % 256 == 0 -> each block maps to exactly one pair
  __shared__ unsigned scnt, spany, stany;
  if (threadIdx.x == 0) { scnt = 0u; spany = 0u; stany = 0u; }
  __syncthreads();
  if (bp) { atomicAdd(&scnt, (unsigned)__popc(bp)); atomicOr(&spany, 1u); }
  if (bt) atomicOr(&stany, 1u);
  __syncthreads();
  if (threadIdx.x == 0) {
    if (scnt)  atomicAdd(&accu[ACNT + pair], scnt);
    if (spany) atomicOr(&accu[APANY + pair], 1u);
    if (stany) atomicOr(&accu[ATANY + pair], 1u);
  }
}

// K4: EDT pass along W directly from boundary bits (seed 0 / INF + min-plus).
// grid = NPAIR * D * (H/8) blocks; block = 8 lines x 32 lanes.
__global__ void k_edt_w(const unsigned* __restrict__ tmbd, float* __restrict__ f) {
  const int blk  = blockIdx.x;
  const int pair = blk / (Dd * (Hh / 8));
  const int rem  = blk % (Dd * (Hh / 8));
  const int z = rem / (Hh / 8), yc = rem % (Hh / 8);
  const int ly = threadIdx.x >> 5, wx = threadIdx.x & 31;
  const int y = yc * 8 + ly;

  __shared__ unsigned wd[8][3];
  if (wx < 3) wd[ly][wx] = tmbd[(size_t)pair * WPP + (size_t)(z * Hh + y) * 3 + wx];
  __syncthreads();

  const int i0 = wx, i1 = wx + 32, i2 = wx + 64;
  int b0 = 1 << 30, b1 = 1 << 30, b2 = 1 << 30;
  auto upd = [&](int j) {
    int d0 = i0 - j; b0 = min(b0, d0 * d0);
    int d1 = i1 - j; b1 = min(b1, d1 * d1);
    int d2 = i2 - j; b2 = min(b2, d2 * d2);
  };
  unsigned u;
  u = wd[ly][0]; for (int j = 0;  j < 32; ++j, u >>= 1) if (u & 1u) upd(j);
  u = wd[ly][1]; for (int j = 32; j < 64; ++j, u >>= 1) if (u & 1u) upd(j);
  u = wd[ly][2]; for (int j = 64; j < 96; ++j, u >>= 1) if (u & 1u) upd(j);

  const size_t base = (size_t)pair * DHW + (size_t)(z * Hh + y) * Ww;
  f[base + i0] = (b0 < (1 << 30)) ? (float)b0 : INFF;
  f[base + i1] = (b1 < (1 << 30)) ? (float)b1 : INFF;
  f[base + i2] = (b2 < (1 << 30)) ? (float)b2 : INFF;
}

// K5: EDT pass along Y. grid = NPAIR*D*3 ; block = 32 wx x 8 ty.
// 96x32 LDS tile staged with CDNA5 async global->LDS copies (ASYNCcnt).
__global__ void k_edt_y(float* __restrict__ f) {
  const int blk  = blockIdx.x;
  const int pair = blk / (Dd * 3);
  const int rem  = blk % (Dd * 3);
  const int z = rem / 3, wc = rem % 3;
  const int wx = threadIdx.x & 31, ty = threadIdx.x >> 5;
  __shared__ float tile[96][32];
  const size_t base = (size_t)pair * DHW + (size_t)z * Hh * Ww + wc * 32 + wx;
  const unsigned lds0 = (unsigned)(uintptr_t)&tile[0][0];  // low 32b = LDS addr
  for (int y = ty; y < Hh; y += 8)
    async_load_lds_b32(lds0 + (unsigned)((y * 32 + wx) * 4),
                       &f[base + (size_t)y * Ww]);
  wait_asynccnt0();
  __syncthreads();
  for (int yo = ty; yo < Hh; yo += 8) {
    float best = tile[0][wx] + (float)(yo * yo);
    for (int j = 1; j < 96; ++j) {
      const int d = yo - j;
      best = fminf(best, tile[j][wx] + (float)(d * d));
    }
    f[base + (size_t)yo * Ww] = best;
  }
}

// K6: EDT pass along Z. grid = NPAIR*H*3 ; same tiling with z-stride H*W.
__global__ void k_edt_z(float* __restrict__ f) {
  const int blk  = blockIdx.x;
  const int pair = blk / (Hh * 3);
  const int rem  = blk % (Hh * 3);
  const int y = rem / 3, wc = rem % 3;
  const int wx = threadIdx.x & 31, tz = threadIdx.x >> 5;
  __shared__ float tile[96][32];
  const size_t base = (size_t)pair * DHW + (size_t)y * Ww + wc * 32 + wx;
  const unsigned lds0 = (unsigned)(uintptr_t)&tile[0][0];
  for (int z = tz; z < Dd; z += 8)
    async_load_lds_b32(lds0 + (unsigned)((z * 32 + wx) * 4),
                       &f[base + (size_t)z * Hh * Ww]);
  wait_asynccnt0();
  __syncthreads();
  for (int zo = tz; zo < Dd; zo += 8) {
    float best = tile[0][wx] + (float)(zo * zo);
    for (int j = 1; j < 96; ++j) {
      const int d = zo - j;
      best = fminf(best, tile[j][wx] + (float)(d * d));
    }
    f[base + (size_t)zo * Hh * Ww] = best;
  }
}

// K7: sum sqrt(dist) over pred-boundary voxels -> per-block partials.
__global__ void k_surf(const unsigned* __restrict__ pmbd,
                       const float* __restrict__ f,
                       float* __restrict__ partials2) {
  const int pair = blockIdx.x / SBLK;
  const int bs   = blockIdx.x % SBLK;
  const int s    = bs * 256 + threadIdx.x;
  const unsigned wbits = pmbd[(size_t)pair * WPP + (s >> 5)];
  const float v = ((wbits >> (s & 31)) & 1u)
                      ? sqrtf(f[(size_t)pair * DHW + s]) : 0.0f;
  const float wsum = wave_sum(v);
  __shared__ float sl[8];
  if ((threadIdx.x & 31) == 0) sl[threadIdx.x >> 5] = wsum;
  __syncthreads();
  if (threadIdx.x == 0) {
    float tot = 0.0f;
    for (int j = 0; j < 8; ++j) tot += sl[j];
    partials2[blockIdx.x] = tot;
  }
}

// K8: reduce surface partials per pair.
__global__ void k_reduce_surf(const float* __restrict__ partials2,
                              float* __restrict__ acc) {
  const int pair = blockIdx.x;
  float s = 0.0f;
  for (int i = threadIdx.x; i < SBLK; i += 256)
    s += partials2[(size_t)pair * SBLK + i];
  const float wsum = wave_sum(s);
  __shared__ float sl[8];
  if ((threadIdx.x & 31) == 0) sl[threadIdx.x >> 5] = wsum;
  __syncthreads();
  if (threadIdx.x == 0) {
    float tot = 0.0f;
    for (int j = 0; j < 8; ++j) tot += sl[j];
    acc[ASURF + pair] = tot;
  }
}

// K9: final scalar combine -> 16 output floats.
__global__ void k_combine(const float* __restrict__ acc, float* __restrict__ out) {
  const unsigned* accu = (const unsigned*)acc;
  const float cw[5] = {1.0f, 2.0f, 1.5f, 2.5f, 1.5f};
  const float bv[5] = {0.0f, 2000.0f, 8000.0f, 1000.0f, 3000.0f};

  float eff[2][5], Wd[2][5];
  for (int b = 0; b < 2; ++b)
    for (int c = 0; c < 5; ++c) {
      const float vols = acc[AVOLS + b * 5 + c];
      const float mult = fminf(sqrtf(bv[c] / fmaxf(vols, EPSF)), 2.5f);
      eff[b][c] = cw[c] * mult;
    }
  for (int c = 0; c < 5; ++c) {
    const float den = fmaxf(fmaxf(eff[0][c], eff[1][c]), EPSF);
    Wd[0][c] = eff[0][c] / den;
    Wd[1][c] = eff[1][c] / den;
  }

  float dice_sum = 0.0f, tv_sum = 0.0f;
  for (int b = 0; b < 2; ++b)
    for (int cp = 0; cp < 4; ++cp) {
      const float inter = acc[AINTER + b * 4 + cp];
      const float psum  = acc[APSUM + b * 4 + cp];
      const float tsum  = acc[AVOLS + b * 5 + cp + 1];
      dice_sum += 1.0f - (2.0f * inter + EPSF) / (psum + tsum + EPSF);
      const float fp = ALPHA * (psum - inter);
      const float fn = BETA * (tsum - inter);
      tv_sum += 1.0f - (inter + EPSF) / (inter + fp + fn + EPSF);
    }
  const float ce      = (acc[ACE] + acc[ACE + 1]) / (float)(Bb * DHW);
  const float dice_ce = 0.5f * (dice_sum / 8.0f) + 0.5f * ce;
  const float tversky = tv_sum / 8.0f;

  float wdv[2], wtv[2], wsf[2], total[2];
  for (int b = 0; b < 2; ++b) {
    float wsum = 0.0f;
    for (int c = 0; c < 5; ++c) wsum += Wd[b][c];
    wdv[b] = dice_ce * wsum;
    wtv[b] = tversky * wsum;
    float sv = 0.0f;
    for (int cp = 0; cp < 4; ++cp) {
      const int pair = b * 4 + cp;
      const unsigned pany = accu[APANY + pair], tany = accu[ATANY + pair];
      float v = acc[ASURF + pair] / fmaxf((float)accu[ACNT + pair], 1.0f);
      if (!isfinite(v)) v = 10.0f;
      const float val = (pany && tany) ? v : ((pany != tany) ? 10.0f : 0.0f);
      sv += val * cw[cp + 1];
    }
    wsf[b]  = sv / 4.0f;
    total[b] = 0.4f * wdv[b] + 0.4f * wtv[b] + 0.2f * wsf[b];
  }
  out[0] = 0.5f * (total[0] + total[1]);
  out[1] = 0.5f * (wdv[0] + wdv[1]);
  out[2] = 0.5f * (wtv[0] + wtv[1]);
  out[3] = 0.5f * (wsf[0] + wsf[1]);
  for (int b = 0; b < 2; ++b)
    for (int c = 0; c < 5; ++c) out[4 + b * 5 + c] = Wd[b][c];
  out[14] = total[0];
  out[15] = total[1];
}

extern "C" void kernel_launch(void* const* d_in, const int* in_sizes, int n_in,
                              void* d_out, int out_size, void* d_ws, size_t ws_size,
                              hipStream_t stream) {
  (void)in_sizes; (void)n_in; (void)out_size; (void)ws_size;
  const float* pred   = (const float*)d_in[0];
  const float* target = (const float*)d_in[1];
  float* out = (float*)d_out;
  char*  ws  = (char*)d_ws;

  float*    acc  = (float*)(ws + ACC_OFF);
  float*    p1   = (float*)(ws + P1_OFF);
  float*    p2   = (float*)(ws + P2_OFF);
  unsigned* pm   = (unsigned*)(ws + PM_OFF);
  unsigned* tm   = (unsigned*)(ws + TM_OFF);
  unsigned* pmbd = (unsigned*)(ws + PMBD_OFF);
  unsigned* tmbd = (unsigned*)(ws + TMBD_OFF);
  float*    f    = (float*)(ws + F_OFF);

  k_init<<<1, 64, 0, stream>>>(acc);
  k_fused<<<NBLK1, 96, 0, stream>>>(pred, target, pm, tm, p1);
  k_reduce_main<<<2 * 14, 256, 0, stream>>>(p1, acc);
  k_boundary<<<(NPAIR * WPP) / 256, 256, 0, stream>>>(pm, tm, pmbd, tmbd,
                                                      (unsigned*)acc);
  k_edt_w<<<NPAIR * Dd * (Hh / 8), 256, 0, stream>>>(tmbd, f);
  k_edt_y<<<NPAIR * Dd * 3, 256, 0, stream>>>(f);
  k_edt_z<<<NPAIR * Hh * 3, 256, 0, stream>>>(f);
  k_surf<<<NPAIR * SBLK, 256, 0, stream>>>(pmbd, f, p2);
  k_reduce_surf<<<NPAIR, 256, 0, stream>>>(p2, acc);
  k_combine<<<1, 1, 0, stream>>>(acc, out);
}